// EPModel_75110388073098
// MI455X (gfx1250) — compile-verified
//
#include <hip/hip_runtime.h>

// ---------------------------------------------------------------------------
// EP model for MI455X (gfx1250): fp32 WMMA GEMM pipeline, pair-major LDS.
// All matrix math via V_WMMA_F32_16X16X4_F32 (wave32).
// ---------------------------------------------------------------------------

typedef __attribute__((ext_vector_type(2))) float v2f;
typedef __attribute__((ext_vector_type(8))) float v8f;

#define EP_BETA 1e-3f
#define EP_LR   0.5f

constexpr int BM = 128;  // block rows
constexpr int BN = 128;  // block cols
constexpr int BK = 32;   // k step

// ---------------------------------------------------------------------------
// GEMM: D = epilogue(alpha * opA(A) @ opB(B), Cin)
//   TA=false: A stored M x K (ld=K).  TA=true: A stored K x M (ld=M).
//   TB=false: B stored K x N (ld=N).  TB=true: B stored N x K (ld=K).
//   EPI=0: D = alpha*acc ; EPI=1: D = Cin - acc ; EPI=2: D = Cin + acc
// Requires M%128==0, N%128==0, K%32==0 (true for all shapes here).
// 8 waves: wave = 32x64 strip = 2x4 WMMA tiles, A/B frags are single b64 LDS
// loads thanks to the K-pair-major shared layout.
// ---------------------------------------------------------------------------
template <bool TA, bool TB, int EPI>
__global__ void __launch_bounds__(256)
gemm_f32_wmma(const float* __restrict__ A, const float* __restrict__ B,
              const float* __restrict__ Cin, float* __restrict__ D,
              int M, int N, int K, float alpha)
{
    // Asp[kp][m] = { A[2kp][m], A[2kp+1][m] } (logical), same for Bsp over n.
    __shared__ v2f Asp[BK / 2][BM + 2];
    __shared__ v2f Bsp[BK / 2][BN + 2];

    const int tid   = threadIdx.x;
    const int lane  = tid & 31;
    const int wid   = tid >> 5;
    const int waveM = wid & 3;   // 32-row strip
    const int waveN = wid >> 2;  // 64-col strip
    const int g     = lane >> 4; // half-wave: K pair select
    const int ln    = lane & 15;

    const int bm = blockIdx.x * BM;
    const int bn = blockIdx.y * BN;

    v8f acc[2][4] = {};  // 2 (M) x 4 (N) 16x16 f32 tiles

    for (int kt = 0; kt < K; kt += BK) {
        // ---- prefetch next K tile (L2-resident working set; keeps pipe warm)
        if (kt + BK < K) {
            const float* pa = TA ? A + (size_t)(kt + BK + (tid >> 5)) * M + bm + 4 * (tid & 31)
                                 : A + (size_t)(bm + (tid >> 3)) * K + kt + BK + 4 * (tid & 7);
            const float* pb = TB ? B + (size_t)(bn + (tid >> 3)) * K + kt + BK + 4 * (tid & 7)
                                 : B + (size_t)(kt + BK + (tid >> 5)) * N + bn + 4 * (tid & 31);
            __builtin_prefetch(pa, 0, 1);
            __builtin_prefetch(pb, 0, 1);
        }

        // ---- stage A tile (4 float4 per thread) ----
        #pragma unroll
        for (int i = 0; i < 4; ++i) {
            int idx4 = i * 256 + tid;
            if (!TA) {
                int kq = idx4 & 7;   // float4 along K (BK/4 = 8)
                int m  = idx4 >> 3;  // 0..127
                float4 v = *(const float4*)(A + (size_t)(bm + m) * K + kt + 4 * kq);
                v2f p0; p0.x = v.x; p0.y = v.y;
                v2f p1; p1.x = v.z; p1.y = v.w;
                Asp[2 * kq + 0][m] = p0;   // single ds_store_b64
                Asp[2 * kq + 1][m] = p1;
            } else {
                int mq = idx4 & 31;  // float4 along M (BM/4 = 32)
                int k  = idx4 >> 5;  // 0..31
                float4 v = *(const float4*)(A + (size_t)(kt + k) * M + bm + 4 * mq);
                ((float*)&Asp[k >> 1][4 * mq + 0])[k & 1] = v.x;
                ((float*)&Asp[k >> 1][4 * mq + 1])[k & 1] = v.y;
                ((float*)&Asp[k >> 1][4 * mq + 2])[k & 1] = v.z;
                ((float*)&Asp[k >> 1][4 * mq + 3])[k & 1] = v.w;
            }
        }
        // ---- stage B tile (4 float4 per thread) ----
        #pragma unroll
        for (int i = 0; i < 4; ++i) {
            int idx4 = i * 256 + tid;
            if (TB) {
                int kq = idx4 & 7;
                int n  = idx4 >> 3;
                float4 v = *(const float4*)(B + (size_t)(bn + n) * K + kt + 4 * kq);
                v2f p0; p0.x = v.x; p0.y = v.y;
                v2f p1; p1.x = v.z; p1.y = v.w;
                Bsp[2 * kq + 0][n] = p0;
                Bsp[2 * kq + 1][n] = p1;
            } else {
                int nq = idx4 & 31;
                int k  = idx4 >> 5;
                float4 v = *(const float4*)(B + (size_t)(kt + k) * N + bn + 4 * nq);
                ((float*)&Bsp[k >> 1][4 * nq + 0])[k & 1] = v.x;
                ((float*)&Bsp[k >> 1][4 * nq + 1])[k & 1] = v.y;
                ((float*)&Bsp[k >> 1][4 * nq + 2])[k & 1] = v.z;
                ((float*)&Bsp[k >> 1][4 * nq + 3])[k & 1] = v.w;
            }
        }
        __syncthreads();

        // ---- 32x64 wave strip: 6 b64 LDS loads -> 8 WMMA per K-step ----
        #pragma unroll
        for (int k0 = 0; k0 < BK; k0 += 4) {
            const int kp = (k0 >> 1) + g;  // K pair {k0+2g, k0+2g+1}
            v2f a0 = Asp[kp][32 * waveM + ln];
            v2f a1 = Asp[kp][32 * waveM + 16 + ln];
            #pragma unroll
            for (int j = 0; j < 4; ++j) {
                v2f b = Bsp[kp][64 * waveN + 16 * j + ln];
                acc[0][j] = __builtin_amdgcn_wmma_f32_16x16x4_f32(
                    false, a0, false, b, (short)0, acc[0][j], false, false);
                acc[1][j] = __builtin_amdgcn_wmma_f32_16x16x4_f32(
                    false, a1, false, b, (short)0, acc[1][j], false, false);
            }
        }
        __syncthreads();
    }

    // ---- epilogue: C/D layout (VGPR r: lanes 0-15 -> M=r, 16-31 -> M=r+8)
    #pragma unroll
    for (int i2 = 0; i2 < 2; ++i2) {
        #pragma unroll
        for (int j = 0; j < 4; ++j) {
            int col = bn + 64 * waveN + 16 * j + ln;
            #pragma unroll
            for (int r = 0; r < 8; ++r) {
                int row = bm + 32 * waveM + 16 * i2 + r + 8 * g;
                size_t idx = (size_t)row * N + col;
                float v = acc[i2][j][r];
                if (EPI == 0)      D[idx] = alpha * v;
                else if (EPI == 1) D[idx] = Cin[idx] - v;
                else               D[idx] = Cin[idx] + v;
            }
        }
    }
}

// ---------------------------------------------------------------------------
// Elementwise kernels (float4, n divisible by 1024; grid = n/1024, block 256)
// ---------------------------------------------------------------------------
__global__ void __launch_bounds__(256) subk(float* __restrict__ o,
                                            const float* __restrict__ a,
                                            const float* __restrict__ b) {
    size_t i = ((size_t)blockIdx.x * 256 + threadIdx.x) * 4;
    float4 av = *(const float4*)(a + i);
    float4 bv = *(const float4*)(b + i);
    *(float4*)(o + i) = make_float4(av.x - bv.x, av.y - bv.y,
                                    av.z - bv.z, av.w - bv.w);
}

__global__ void __launch_bounds__(256) g3k(float* __restrict__ r3,
                                           const float* __restrict__ t3,
                                           const float* __restrict__ y) {
    size_t i = ((size_t)blockIdx.x * 256 + threadIdx.x) * 4;
    float4 rv = *(const float4*)(r3 + i);
    float4 tv = *(const float4*)(t3 + i);
    float4 yv = *(const float4*)(y + i);
    rv.x += EP_BETA * (tv.x - yv.x);
    rv.y += EP_BETA * (tv.y - yv.y);
    rv.z += EP_BETA * (tv.z - yv.z);
    rv.w += EP_BETA * (tv.w - yv.w);
    *(float4*)(r3 + i) = rv;
}

__global__ void __launch_bounds__(256) updk(float* __restrict__ t,
                                            const float* __restrict__ gr) {
    size_t i = ((size_t)blockIdx.x * 256 + threadIdx.x) * 4;
    float4 tv = *(const float4*)(t + i);
    float4 gv = *(const float4*)(gr + i);
    tv.x -= EP_LR * gv.x; tv.y -= EP_LR * gv.y;
    tv.z -= EP_LR * gv.z; tv.w -= EP_LR * gv.w;
    *(float4*)(t + i) = tv;
}

__global__ void __launch_bounds__(256) copyk(float* __restrict__ dst,
                                             const float* __restrict__ src) {
    size_t i = ((size_t)blockIdx.x * 256 + threadIdx.x) * 4;
    *(float4*)(dst + i) = *(const float4*)(src + i);
}

// ---------------------------------------------------------------------------
// Deterministic two-stage loss reduction: mean((s3 - y)^2) over 1024x512.
// ---------------------------------------------------------------------------
__global__ void __launch_bounds__(256) loss_partial(const float* __restrict__ s,
                                                    const float* __restrict__ y,
                                                    float* __restrict__ part) {
    __shared__ float red[256];
    size_t i = ((size_t)blockIdx.x * 256 + threadIdx.x) * 4;
    float4 a = *(const float4*)(s + i);
    float4 b = *(const float4*)(y + i);
    float dx = a.x - b.x, dy = a.y - b.y, dz = a.z - b.z, dw = a.w - b.w;
    red[threadIdx.x] = dx * dx + dy * dy + dz * dz + dw * dw;
    __syncthreads();
    for (int off = 128; off > 0; off >>= 1) {
        if (threadIdx.x < off) red[threadIdx.x] += red[threadIdx.x + off];
        __syncthreads();
    }
    if (threadIdx.x == 0) part[blockIdx.x] = red[0];
}

__global__ void __launch_bounds__(256) loss_final(const float* __restrict__ part,
                                                  float* __restrict__ out) {
    __shared__ float red[256];
    red[threadIdx.x] = part[threadIdx.x] + part[threadIdx.x + 256];
    __syncthreads();
    for (int off = 128; off > 0; off >>= 1) {
        if (threadIdx.x < off) red[threadIdx.x] += red[threadIdx.x + off];
        __syncthreads();
    }
    if (threadIdx.x == 0) out[0] = red[0] * (1.0f / 524288.0f);
}

// ---------------------------------------------------------------------------
// Orchestration
// ---------------------------------------------------------------------------
#define GEMM(TA, TB, EPI, M, N, K, Ap, Bp, Cp, Dp, alpha)                     \
    gemm_f32_wmma<TA, TB, EPI><<<dim3((M) / BM, (N) / BN), 256, 0, stream>>>( \
        Ap, Bp, Cp, Dp, M, N, K, alpha)

#define EW(kernel, n, ...) kernel<<<dim3((n) / 1024), 256, 0, stream>>>(__VA_ARGS__)

extern "C" void kernel_launch(void* const* d_in, const int* in_sizes, int n_in,
                              void* d_out, int out_size, void* d_ws, size_t ws_size,
                              hipStream_t stream)
{
    const float* x  = (const float*)d_in[0];  // 1024 x 1024
    const float* y  = (const float*)d_in[1];  // 1024 x 512
    const float* W0 = (const float*)d_in[2];  // 1024 x 1024
    const float* W1 = (const float*)d_in[3];  // 1024 x 1024
    const float* W2 = (const float*)d_in[4];  // 1024 x 512
    float* out = (float*)d_out;               // 2*1048576 + 524288 + 1 floats

    const int Bt = 1024;                      // batch
    const int D0 = 1024, D1 = 1024, D2 = 1024, D3 = 512;
    const size_t n1 = (size_t)Bt * D1;        // 1048576
    const size_t n2 = (size_t)Bt * D2;        // 1048576
    const size_t n3 = (size_t)Bt * D3;        // 524288

    float* ws = (float*)d_ws;
    float* p0 = ws;            // x @ W0 (constant through relaxation)
    float* t1 = p0 + n1;       // state s1
    float* t2 = t1 + n1;       // state s2
    float* t3 = t2 + n2;       // state s3
    float* r1 = t3 + n3;       // residual / gradient g1
    float* r2 = r1 + n1;       // residual / gradient g2
    float* r3 = r2 + n2;       // residual / gradient g3
    float* part = r3 + n3;     // 512 loss partials

    // ---- free phase (exact forward chain) + loss -------------------------
    GEMM(false, false, 0, Bt, D1, D0, x,  W0, (const float*)nullptr, p0, 1.0f);
    EW(copyk, n1, t1, p0);
    GEMM(false, false, 0, Bt, D2, D1, t1, W1, (const float*)nullptr, t2, 1.0f);
    GEMM(false, false, 0, Bt, D3, D2, t2, W2, (const float*)nullptr, t3, 1.0f);
    loss_partial<<<512, 256, 0, stream>>>(t3, y, part);
    loss_final<<<1, 256, 0, stream>>>(part, out + 2621440);

    // ---- clamped relaxation: 20 steps ------------------------------------
    for (int it = 0; it < 20; ++it) {
        // fused residuals: r2 = t2 - t1@W1 ; r3 = t3 - t2@W2
        GEMM(false, false, 1, Bt, D2, D1, t1, W1, t2, r2, 1.0f);
        GEMM(false, false, 1, Bt, D3, D2, t2, W2, t3, r3, 1.0f);
        EW(subk, n1, r1, t1, p0);   // r1 = t1 - x@W0
        // g1 = r1 - r2 @ W1^T   (in place on r1; uses raw r2)
        GEMM(false, true, 1, Bt, D1, D2, r2, W1, r1, r1, 1.0f);
        // g2 = r2 - r3 @ W2^T   (in place on r2; uses raw r3)
        GEMM(false, true, 1, Bt, D2, D3, r3, W2, r2, r2, 1.0f);
        // g3 = r3 + beta*(t3 - y)
        EW(g3k, n3, r3, t3, y);
        if (it == 0) {  // coupled first step: backprop chain along states
            GEMM(false, true, 2, Bt, D2, D3, r3, W2, r2, r2, 1.0f);  // g2 += g3@W2^T
            GEMM(false, true, 2, Bt, D1, D2, r2, W1, r1, r1, 1.0f);  // g1 += g2@W1^T
        }
        EW(updk, n1, t1, r1);
        EW(updk, n2, t2, r2);
        EW(updk, n3, t3, r3);
    }

    // ---- EP weight grads: gW_i = (1/beta) s_i^T (s_i W_i - s_{i+1})
    //      = (-1/beta) s_i^T r_{i+1}  with r fused as above -----------------
    GEMM(false, false, 1, Bt, D2, D1, t1, W1, t2, r2, 1.0f);  // r2 = t2 - t1@W1
    GEMM(false, false, 1, Bt, D3, D2, t2, W2, t3, r3, 1.0f);  // r3 = t3 - t2@W2
    EW(subk, n1, r1, t1, p0);                                 // r1 = t1 - x@W0
    GEMM(true, false, 0, D0, D1, Bt, x,  r1, (const float*)nullptr, out,           -1.0f / EP_BETA);
    GEMM(true, false, 0, D1, D2, Bt, t1, r2, (const float*)nullptr, out + 1048576, -1.0f / EP_BETA);
    GEMM(true, false, 0, D2, D3, Bt, t2, r3, (const float*)nullptr, out + 2097152, -1.0f / EP_BETA);
}